// Graph3DBiasPBC_15616501088376
// MI455X (gfx1250) — compile-verified
//
#include <hip/hip_runtime.h>
#include <hip/hip_bf16.h>

// ---------------- static problem configuration (mirrors reference) ----------
#define BATCH   32
#define NATOMS  128
#define NCELLS  25            // 5*5*1 periodic images (z pbc off)
#define NCAND   (NATOMS * NCELLS)   // 3200 candidates per center atom
#define MAXNB   32            // max_num_neighbors_threshold
#define R2      64.0f         // RADIUS^2

typedef float v2f __attribute__((ext_vector_type(2)));
typedef float v8f __attribute__((ext_vector_type(8)));

// ---------------------------------------------------------------------------
// Kernel 1: per-center-atom stable top-32 selection (1 wave32 per atom).
// Produces, per atom: threshold = bits of 32nd-smallest within-d2 (0xFFFFFFFF
// if <=32 within), and cut_idx = candidate index of last kept tie (stable
// argsort semantics), plus atomicAdd of clamped neighbor counts per image.
// ---------------------------------------------------------------------------
__global__ __launch_bounds__(32) void knn_select_kernel(
    const float* __restrict__ pos, const float* __restrict__ cell,
    int* __restrict__ nbimg, unsigned* __restrict__ tarr,
    unsigned* __restrict__ cutarr)
{
    __shared__ unsigned sbits[NCAND];           // 12.8 KB

    const int lane = threadIdx.x;
    const int b    = blockIdx.x / NATOMS;
    const int i    = blockIdx.x % NATOMS;

    const float* cb = cell + b * 9;             // row-vector cell convention
    const float c00 = cb[0], c01 = cb[1], c02 = cb[2];
    const float c10 = cb[3], c11 = cb[4], c12 = cb[5];
    const float* pb = pos + (size_t)b * NATOMS * 3;
    const float pix = pb[i*3+0], piy = pb[i*3+1], piz = pb[i*3+2];

    int nwi = 0;
    for (int cand = lane; cand < NCAND; cand += 32) {
        int j  = cand / NCELLS;
        int cc = cand % NCELLS;                  // cc = ixi*5 + iyi (meshgrid ij)
        float fx = (float)(cc / 5 - 2);
        float fy = (float)(cc % 5 - 2);
        float qx = pb[j*3+0] + fx*c00 + fy*c10;
        float qy = pb[j*3+1] + fx*c01 + fy*c11;
        float qz = pb[j*3+2] + fx*c02 + fy*c12;
        float dx = qx - pix, dy = qy - piy, dz = qz - piz;
        float d2 = dx*dx + dy*dy + dz*dz;
        bool within = (d2 <= R2) & (d2 > 1e-4f);
        sbits[cand] = within ? __float_as_uint(d2) : 0xFFFFFFFFu;
        nwi += within ? 1 : 0;
    }
    __syncthreads();

    // wave32 reduce of within-count (full butterfly -> all lanes hold sum)
    for (int o = 16; o > 0; o >>= 1) nwi += __shfl_xor(nwi, o, 32);
    if (lane == 0) atomicAdd(&nbimg[b], nwi < MAXNB ? nwi : MAXNB);

    const int gi = blockIdx.x;                   // b*NATOMS + i
    if (nwi <= MAXNB) {                          // keep everything within radius
        if (lane == 0) { tarr[gi] = 0xFFFFFFFFu; cutarr[gi] = 0xFFFFFFFFu; }
        return;
    }

    // exact radix select of the 32nd-smallest key (nonneg float bits are
    // order-preserving as u32; invalid = 0xFFFFFFFF sorts last).
    unsigned prefix = 0u;
    int k = MAXNB;                               // 1-based target rank
    for (int bit = 31; bit >= 0; --bit) {
        const unsigned pfx = prefix >> bit;      // current prefix with bit=0
        int c0 = 0;
        for (int cand = lane; cand < NCAND; cand += 32)
            c0 += ((sbits[cand] >> bit) == pfx) ? 1 : 0;
        for (int o = 16; o > 0; o >>= 1) c0 += __shfl_xor(c0, o, 32);
        if (k > c0) { k -= c0; prefix |= (1u << bit); }
    }
    // prefix = 32nd smallest value; k = #ties (==prefix) to keep, in index order
    unsigned cum = 0, cutv = 0xFFFFFFFFu;
    for (int base = 0; base < NCAND; base += 32) {
        bool pred = (sbits[base + lane] == prefix);
        unsigned long long bal = __ballot(pred);
        unsigned cnt = (unsigned)__popcll(bal);
        if (cum + cnt >= (unsigned)k) {
            unsigned need = (unsigned)k - cum;   // 1-based within this block
            for (unsigned nn = 1; nn < need; ++nn) bal &= (bal - 1ull);
            cutv = (unsigned)base + (unsigned)(__ffsll(bal) - 1);
            break;
        }
        cum += cnt;
    }
    if (lane == 0) { tarr[gi] = prefix; cutarr[gi] = cutv; }
}

// ---------------------------------------------------------------------------
// Kernel 2: WMMA d2 tiles + branchless output streaming with NT stores.
// d2 = P' Q'^T + |p_i|^2 with P'=(-2p,1), Q'=(q,|q|^2): one
// v_wmma_f32_16x16x4_f32 per 16x16 (center x candidate) tile.
// Grid: one block per (b, center-tile, candidate-tile group); its 8 waves cover 8 candidate tiles.
// ---------------------------------------------------------------------------
__global__ __launch_bounds__(256) void wmma_edges_kernel(
    const float* __restrict__ pos, const float* __restrict__ cell,
    const int* __restrict__ nbimg, const unsigned* __restrict__ tarr,
    const unsigned* __restrict__ cutarr,
    float* __restrict__ dist_out, int* __restrict__ ctype_out,
    unsigned char* __restrict__ mask_out, int* __restrict__ cnt_out)
{
    __shared__ float    a4[16][4];   // P' rows for the 16 centers of this tile
    __shared__ float    p2s[16];     // |p_i|^2
    __shared__ unsigned tsh[16], cuts[16];
    __shared__ float    cellf[6];    // cell rows 0 and 1

    const int tid  = threadIdx.x;
    const int b    = blockIdx.x / 200;   // 8 center-tiles * 25 wave-groups
    const int rem  = blockIdx.x % 200;
    const int it   = rem / 25;           // center tile [0,8)
    const int jblk = rem % 25;           // candidate tile group [0,25)

    const float* pb = pos + (size_t)b * NATOMS * 3;
    if (tid < 16) {
        int m = it*16 + tid;
        float px = pb[m*3+0], py = pb[m*3+1], pz = pb[m*3+2];
        a4[tid][0] = -2.0f*px; a4[tid][1] = -2.0f*py;
        a4[tid][2] = -2.0f*pz; a4[tid][3] = 1.0f;
        p2s[tid]  = px*px + py*py + pz*pz;
        tsh[tid]  = tarr[b*NATOMS + m];
        cuts[tid] = cutarr[b*NATOMS + m];
    } else if (tid < 22) {
        cellf[tid - 16] = cell[b*9 + (tid - 16)];
    }
    __syncthreads();

    const int w    = tid >> 5;
    const int lane = tid & 31;
    const int l16  = lane & 15;
    const int hi   = lane >> 4;                // half-wave selects K in {0,1} vs {2,3}
    const int jc0  = (jblk*8 + w) * 16;
    const int cand = jc0 + l16;                // candidate = j*25 + c
    const int j    = cand / 25;
    const int cc   = cand % 25;
    const int ixi  = cc / 5, iyi = cc % 5;
    const float fx = (float)(ixi - 2), fy = (float)(iyi - 2);

    float qx = pb[j*3+0] + fx*cellf[0] + fy*cellf[3];
    float qy = pb[j*3+1] + fx*cellf[1] + fy*cellf[4];
    float qz = pb[j*3+2] + fx*cellf[2] + fy*cellf[5];
    float q2 = qx*qx + qy*qy + qz*qz;

    // Hoist per-M LDS operands for this half-wave (M = r + 8*hi, r = 0..7)
    // so the unrolled output loop is pure VALU + stores (no ds under branches).
    float    p2r[8];
    unsigned tr[8], cr[8];
#pragma unroll
    for (int r = 0; r < 8; ++r) {
        const int M = r + 8*hi;
        p2r[r] = p2s[M];
        tr[r]  = tsh[M];
        cr[r]  = cuts[M];
    }

    // A 16x4 layout: lanes 0-15 = M, VGPR0/1 = K {0,1} (lo half) / {2,3} (hi half)
    v2f a, bf;
    a.x = a4[l16][2*hi];
    a.y = a4[l16][2*hi + 1];
    // B 4x16 mirrored: lanes = N, VGPR0/1 = K {0,1} / {2,3}
    bf.x = hi ? qz : qx;
    bf.y = hi ? q2 : qy;

    v8f cacc = {};
    // D = A x B + 0  (EXEC all ones: 256 fully-active threads, no divergence here)
    v8f d = __builtin_amdgcn_wmma_f32_16x16x4_f32(
        /*neg_a=*/false, a, /*neg_b=*/false, bf,
        /*c_mod=*/(short)0, cacc, /*reuse_a=*/false, /*reuse_b=*/false);

    // reference quirk: mults = [nr1*nr2, nr1, 1] = [5,5,1] -> ct = 5*(ixi+iyi)
    const int ctval = 5 * (ixi + iyi);

    // Streaming, write-once outputs: use non-temporal stores so 118 MB of
    // output doesn't wash the caches that hold pos/cell/thresholds.
    size_t eidx = ((size_t)(b*NATOMS + it*16 + 8*hi)) * NCAND + (size_t)cand;
#pragma unroll
    for (int r = 0; r < 8; ++r) {
        float d2 = d[r] + p2r[r];
        unsigned ub = __float_as_uint(d2);
        // branchless predicate: bitwise ops, no short-circuit exec branches
        bool within = (d2 <= R2) & (d2 > 1e-4f);
        bool keep   = within & (((ub < tr[r]) | ((ub == tr[r]) & ((unsigned)cand <= cr[r]))));
        float dist  = keep ? sqrtf(d2) : 0.0f;
        int   ctv   = keep ? ctval : 0;
        __builtin_nontemporal_store(dist, dist_out + eidx);
        __builtin_nontemporal_store(ctv,  ctype_out + eidx);
        __builtin_nontemporal_store((unsigned char)(keep ? 1 : 0), mask_out + eidx);
        eidx += NCAND;                          // next M row
    }

    if (blockIdx.x == 0 && tid < BATCH) cnt_out[tid] = nbimg[tid];
}

// ---------------------------------------------------------------------------
extern "C" void kernel_launch(void* const* d_in, const int* in_sizes, int n_in,
                              void* d_out, int out_size, void* d_ws, size_t ws_size,
                              hipStream_t stream) {
    (void)in_sizes; (void)n_in; (void)out_size; (void)ws_size;
    const float* pos  = (const float*)d_in[0];   // [B, N, 3]
    const float* cell = (const float*)d_in[1];   // [B, 3, 3]

    // workspace layout (~33 KB)
    int*      nbimg  = (int*)d_ws;                               // [B]
    unsigned* tarr   = (unsigned*)((char*)d_ws + 256);           // [B*N]
    unsigned* cutarr = tarr + BATCH * NATOMS;                    // [B*N]

    // output layout: dist f32 [E] | cell_type i32 [E] | mask u8 [E] | counts i32 [B]
    const size_t E = (size_t)BATCH * NATOMS * NCAND;             // 13,107,200
    float*         dist  = (float*)d_out;
    int*           ctype = (int*)d_out + E;
    unsigned char* mask  = (unsigned char*)((float*)d_out + 2*E);
    int*           cnts  = (int*)(mask + E);                     // 9E bytes in, 4B aligned

    hipMemsetAsync(nbimg, 0, BATCH * sizeof(int), stream);
    knn_select_kernel<<<BATCH * NATOMS, 32, 0, stream>>>(pos, cell, nbimg, tarr, cutarr);
    wmma_edges_kernel<<<BATCH * 8 * 25, 256, 0, stream>>>(pos, cell, nbimg, tarr, cutarr,
                                                          dist, ctype, mask, cnts);
}